// TorchModel_69741678952700
// MI455X (gfx1250) — compile-verified
//
#include <hip/hip_runtime.h>

// combine_sec = einsum('se,sc->sec', g1[:,None]*mask1, loc1)
//             + einsum('se,sc->sec', g2[:,None]*mask2, loc2)
// Per token s this is a rank-2 outer product: a 16(E) x 512(C) tile computed
// as 32 WMMA F32 16x16x4 tiles with K=2 live (K=2,3 zeroed via lane select).
//
// S=4096, E=16, C=512. Output 128 MiB f32 -> pure store-bandwidth bound
// (~151 MB total traffic / 23.3 TB/s ~= 6.5 us floor).

#define S_TOK 4096
#define E_DIM 16
#define C_DIM 512

typedef __attribute__((ext_vector_type(2))) float v2f;
typedef __attribute__((ext_vector_type(8))) float v8f;

__global__ __launch_bounds__(256) void moe_combine_wmma(
    const float* __restrict__ gates1_s,
    const float* __restrict__ loc1,      // [S, C]
    const float* __restrict__ gates2_s,
    const float* __restrict__ loc2,      // [S, C]
    const float* __restrict__ mask1,     // [S, E]
    const float* __restrict__ mask2,     // [S, E]
    float* __restrict__ out)             // [S, E, C]
{
    const int s    = blockIdx.x;          // one token per block
    const int tid  = threadIdx.x;         // 256 threads = 8 waves (wave32)
    const int wave = tid >> 5;            // 0..7
    const int lane = tid & 31;
    const int n    = lane & 15;           // col within 16-wide tile / row index e
    const bool lo  = (lane < 16);         // lanes 0-15 carry K=0,1; 16-31 carry K=2,3

    // Lane select instead of EXEC-masked loads: avoids saveexec/branch per tile.
    const float sel = lo ? 1.0f : 0.0f;

    // Block-uniform gate scalars -> scalar loads.
    const float g1 = gates1_s[s];
    const float g2 = gates2_s[s];

    // A matrix 16x4 (ExK): K0 = g1*mask1[e], K1 = g2*mask2[e], K2=K3=0.
    // Lanes 16-31 (which would carry K=2,3) are zeroed by sel.
    const size_t mrow = (size_t)s * E_DIM;
    const float m1v = mask1[mrow + n];    // all lanes load (dup addrs coalesce)
    const float m2v = mask2[mrow + n];
    v2f a;
    a.x = (g1 * m1v) * sel;
    a.y = (g2 * m2v) * sel;

    const float* L1row = loc1 + (size_t)s * C_DIM;
    const float* L2row = loc2 + (size_t)s * C_DIM;
    float* outtok      = out  + (size_t)s * E_DIM * C_DIM;

    // Prefetch all 4 B tiles (L1/L2 slices) as one load clause.
    float b1[4], b2[4];
#pragma unroll
    for (int i = 0; i < 4; ++i) {
        const int cbase = (wave * 4 + i) * 16;
        b1[i] = L1row[cbase + n];         // unconditional, all lanes
        b2[i] = L2row[cbase + n];
    }

    // D layout: lanes 0-15 hold rows M=0..7 (vgpr v -> M=v), lanes 16-31 rows M=8..15.
    const int mbase = lo ? 0 : 8;

#pragma unroll
    for (int i = 0; i < 4; ++i) {
        const int cbase = (wave * 4 + i) * 16;

        // B matrix 4x16 (KxC): K0 = loc1 slice, K1 = loc2 slice, K2=K3=0 via sel.
        v2f b;
        b.x = b1[i] * sel;
        b.y = b2[i] * sel;

        v8f c = {};  // accumulate from zero
        // 8-arg form: (neg_a, A, neg_b, B, c_mod, C, reuse_a, reuse_b)
        v8f d = __builtin_amdgcn_wmma_f32_16x16x4_f32(
            false, a, false, b, (short)0, c, false, false);

        // Lane writes 8 rows (stride C_DIM); across lanes each row is a
        // contiguous 64B segment, and consecutive tiles keep rows contiguous
        // so writes combine into full cachelines in L2.
        float* p = outtok + (size_t)mbase * C_DIM + cbase + n;
#pragma unroll
        for (int v = 0; v < 8; ++v) {
            p[(size_t)v * C_DIM] = d[v];
        }
    }
}

extern "C" void kernel_launch(void* const* d_in, const int* in_sizes, int n_in,
                              void* d_out, int out_size, void* d_ws, size_t ws_size,
                              hipStream_t stream)
{
    const float* gates1 = (const float*)d_in[0];
    const float* loc1   = (const float*)d_in[1];
    const float* gates2 = (const float*)d_in[2];
    const float* loc2   = (const float*)d_in[3];
    const float* mask1  = (const float*)d_in[4];
    const float* mask2  = (const float*)d_in[5];
    float* out          = (float*)d_out;

    dim3 grid(S_TOK);
    dim3 block(256);
    moe_combine_wmma<<<grid, block, 0, stream>>>(gates1, loc1, gates2, loc2,
                                                 mask1, mask2, out);
}